// XRisk_85469849190438
// MI455X (gfx1250) — compile-verified
//
#include <hip/hip_runtime.h>
#include <math.h>

typedef float v4f __attribute__((ext_vector_type(4)));
typedef int   v4i __attribute__((vector_size(16)));
typedef __attribute__((address_space(1))) v4i v4i_gbl;   // global-AS int4
typedef __attribute__((address_space(3))) v4i v4i_lds;   // LDS-AS int4

#define T_MAX   10000
#define BATCH   4096
#define CHUNK_T 40
#define NCHUNK  (T_MAX / CHUNK_T)                 // 250
#define TB      4                                 // tile rows per async buffer
#define NTILES  (CHUNK_T / TB)                    // 10
#define BLOCKA  1024
#define ITEMS   ((T_MAX + BLOCKA - 1) / BLOCKA)   // 10

// Constants (double-precision host math, truncated to f32)
#define LOG_GAMMA   0.019802627296179713f      // log(1.02)
#define ALPHA_C     (-1.0050335853501442e-05f) // log(0.99)/1000
#define INIT_LOGW   (-1.3944604627209042f)     // log(0.248)
#define INIT_LOGP   (-6.907755278982137f)      // log(0.001)
#define INV_TEMP    (1.0f / 1000.0f)
#define EPS_C       1e-12f

// ws layout
#define OFF_ELOGIT  0           // T_MAX floats: exp(logit_t)
#define OFF_PREFW   40064       // (T_MAX+1) floats: exclusive prefix of w_t
#define OFF_FD      81920       // NCHUNK*BATCH bytes: chunk-local first-death index

#define HAS_ASYNC_LDS (__has_builtin(__builtin_amdgcn_global_load_async_to_lds_b128) && \
                       __has_builtin(__builtin_amdgcn_s_wait_asynccnt))

// ---------------------------------------------------------------------------
// Block-wide exclusive prefix sum (Hillis-Steele, shifted form: safe with -inf)
__device__ __forceinline__ float block_scan_excl(float v, float* sc, int tid, float* totOut) {
  sc[tid] = v;
  __syncthreads();
  #pragma unroll
  for (int off = 1; off < BLOCKA; off <<= 1) {
    float add = (tid >= off) ? sc[tid - off] : 0.0f;
    __syncthreads();
    sc[tid] += add;
    __syncthreads();
  }
  float excl = (tid == 0) ? 0.0f : sc[tid - 1];
  float tot  = sc[BLOCKA - 1];
  __syncthreads();
  if (totOut) *totOut = tot;
  return excl;
}

// Kernel A: batch-independent scalar recurrences via chained parallel prefix sums.
__global__ __launch_bounds__(BLOCKA)
void xrisk_precompute(const float* __restrict__ acts,
                      float* __restrict__ elogit,
                      float* __restrict__ prefW) {
  __shared__ float sc[BLOCKA];
  const int tid  = threadIdx.x;
  const int base = tid * ITEMS;

  float fa[ITEMS], tw[ITEMS];
  float sw = 0.0f;
  #pragma unroll
  for (int i = 0; i < ITEMS; ++i) {
    int t = base + i;
    if (t < T_MAX) {
      fa[i] = expf(acts[t]);                       // f = exp(a)
      tw[i] = LOG_GAMMA + logf(1.0f - fa[i]);      // per-step logw increment
    } else { fa[i] = 0.0f; tw[i] = 0.0f; }
    sw += tw[i];
  }

  // prefix of logw increments -> logw carried into each step
  float excl_w = block_scan_excl(sw, sc, tid, nullptr);
  float lw = INIT_LOGW + excl_w;

  float tp[ITEMS], wl[ITEMS];
  float sp = 0.0f, swl = 0.0f;
  #pragma unroll
  for (int i = 0; i < ITEMS; ++i) {
    int t = base + i;
    if (t < T_MAX) {
      tp[i] = ALPHA_C * fa[i] * expf(lw);  // logp increment (OLD logw; -inf past overflow, like JAX)
      lw += tw[i];                         // logw_after
      wl[i] = expf(lw * INV_TEMP);         // reward weight w_t
    } else { tp[i] = 0.0f; wl[i] = 0.0f; }
    sp  += tp[i];
    swl += wl[i];
  }

  // prefix of logp increments -> logp carried into each step
  float excl_p = block_scan_excl(sp, sc, tid, nullptr);
  float lp = INIT_LOGP + excl_p;
  #pragma unroll
  for (int i = 0; i < ITEMS; ++i) {
    int t = base + i;
    if (t < T_MAX) {
      float p = expf(lp);                  // exp(-inf)=0 -> odds 0 -> never dies
      elogit[t] = p / (1.0f - p);          // exp(logit) = odds ratio
      lp += tp[i];
    }
  }

  // prefix of w_t -> cumrew lookup table (exclusive)
  float tot;
  float excl_wl = block_scan_excl(swl, sc, tid, &tot);
  float c = excl_wl;
  #pragma unroll
  for (int i = 0; i < ITEMS; ++i) {
    int t = base + i;
    if (t < T_MAX) { prefW[t] = c; c += wl[i]; }
  }
  if (tid == BLOCKA - 1) prefW[T_MAX] = tot;
}

// ---------------------------------------------------------------------------
// Kernel B: HBM-bound streaming pass over u1/u2 (328 MB total, read once).
// gfx1250 path: double-buffered GLOBAL_LOAD_ASYNC_TO_LDS_B128 + S_WAIT_ASYNCCNT.
// Each thread owns 4 consecutive batch lanes; each thread's LDS slots are
// private to it, so wave-local asynccnt waits are the only sync needed.
// hard(t,b) <=> log(u2+e) > (elogit_t - e) * log(u1+e)   [log(u1+e) < 0]
__global__ __launch_bounds__(256)
void xrisk_stream(const float* __restrict__ u1,
                  const float* __restrict__ u2,
                  const float* __restrict__ elogit,
                  unsigned int* __restrict__ fd /* packed 4x u8 per quad */) {
  __shared__ float th[CHUNK_T];
  const int tid   = threadIdx.x;
  const int chunk = blockIdx.y;
  const int t0    = chunk * CHUNK_T;
  if (tid < CHUNK_T) th[tid] = elogit[t0 + tid] - EPS_C;
  __syncthreads();

  const int g4 = blockIdx.x * blockDim.x + tid;   // batch quad index [0, 1024)
  int f0 = CHUNK_T, f1 = CHUNK_T, f2 = CHUNK_T, f3 = CHUNK_T;

#if HAS_ASYNC_LDS
  __shared__ v4f s1[2][TB * 256];
  __shared__ v4f s2[2][TB * 256];

  // Issue one TB-row tile of async b128 loads into buffer `bf` (8 ops/thread).
  auto issue_tile = [&](int bf, int tbase) {
    asm volatile("s_wait_dscnt 0" ::: "memory");  // close LDS WAR window on buffer reuse
    #pragma unroll
    for (int r = 0; r < TB; ++r) {
      const float* src1 = u1 + (size_t)(tbase + r) * BATCH + (size_t)g4 * 4;
      __builtin_amdgcn_global_load_async_to_lds_b128(
          (v4i_gbl*)src1, (v4i_lds*)&s1[bf][r * 256 + tid], 0, 0);
    }
    #pragma unroll
    for (int r = 0; r < TB; ++r) {
      const float* src2 = u2 + (size_t)(tbase + r) * BATCH + (size_t)g4 * 4;
      __builtin_amdgcn_global_load_async_to_lds_b128(
          (v4i_gbl*)src2, (v4i_lds*)&s2[bf][r * 256 + tid], 0, 0);
    }
  };

  issue_tile(0, t0);
  int buf = 0;
  for (int tile = 0; tile < NTILES; ++tile) {
    if (tile + 1 < NTILES) {
      issue_tile(buf ^ 1, t0 + (tile + 1) * TB);
      __builtin_amdgcn_s_wait_asynccnt(8);   // current tile retired; next tile in flight
    } else {
      __builtin_amdgcn_s_wait_asynccnt(0);
    }
    #pragma unroll
    for (int r = 0; r < TB; ++r) {
      int tt = tile * TB + r;
      v4f x1 = s1[buf][r * 256 + tid];
      v4f x2 = s2[buf][r * 256 + tid];
      float m = th[tt];
      bool h0 = logf(x2.x + EPS_C) > m * logf(x1.x + EPS_C);
      bool h1 = logf(x2.y + EPS_C) > m * logf(x1.y + EPS_C);
      bool h2 = logf(x2.z + EPS_C) > m * logf(x1.z + EPS_C);
      bool h3 = logf(x2.w + EPS_C) > m * logf(x1.w + EPS_C);
      f0 = (h0 && f0 == CHUNK_T) ? tt : f0;
      f1 = (h1 && f1 == CHUNK_T) ? tt : f1;
      f2 = (h2 && f2 == CHUNK_T) ? tt : f2;
      f3 = (h3 && f3 == CHUNK_T) ? tt : f3;
    }
    buf ^= 1;
  }
#else
  const v4f* __restrict__ p1 = (const v4f*)u1;
  const v4f* __restrict__ p2 = (const v4f*)u2;
  #pragma unroll 2
  for (int tt = 0; tt < CHUNK_T; ++tt) {
    size_t idx = (size_t)(t0 + tt) * (BATCH / 4) + g4;
    v4f x1 = __builtin_nontemporal_load(p1 + idx);
    v4f x2 = __builtin_nontemporal_load(p2 + idx);
    float m = th[tt];
    bool h0 = logf(x2.x + EPS_C) > m * logf(x1.x + EPS_C);
    bool h1 = logf(x2.y + EPS_C) > m * logf(x1.y + EPS_C);
    bool h2 = logf(x2.z + EPS_C) > m * logf(x1.z + EPS_C);
    bool h3 = logf(x2.w + EPS_C) > m * logf(x1.w + EPS_C);
    f0 = (h0 && f0 == CHUNK_T) ? tt : f0;
    f1 = (h1 && f1 == CHUNK_T) ? tt : f1;
    f2 = (h2 && f2 == CHUNK_T) ? tt : f2;
    f3 = (h3 && f3 == CHUNK_T) ? tt : f3;
  }
#endif

  unsigned int packed = (unsigned)f0 | ((unsigned)f1 << 8) |
                        ((unsigned)f2 << 16) | ((unsigned)f3 << 24);
  fd[(size_t)chunk * (BATCH / 4) + g4] = packed;
}

// ---------------------------------------------------------------------------
// Kernel C: resolve global first-death per batch element, gather prefW, mean.
__global__ __launch_bounds__(1024)
void xrisk_reduce(const unsigned char* __restrict__ fd,
                  const float* __restrict__ prefW,
                  float* __restrict__ out) {
  __shared__ float sr[1024];
  const int tid = threadIdx.x;
  float acc = 0.0f;
  for (int b = tid; b < BATCH; b += 1024) {
    int tstar = T_MAX;
    for (int c = 0; c < NCHUNK; ++c) {
      unsigned char v = fd[(size_t)c * BATCH + b];
      if (v < CHUNK_T) { tstar = c * CHUNK_T + (int)v; break; }
    }
    acc += prefW[tstar];
  }
  sr[tid] = acc;
  __syncthreads();
  #pragma unroll
  for (int off = 512; off > 0; off >>= 1) {
    if (tid < off) sr[tid] += sr[tid + off];
    __syncthreads();
  }
  if (tid == 0) out[0] = sr[0] * (1.0f / (float)BATCH);
}

extern "C" void kernel_launch(void* const* d_in, const int* in_sizes, int n_in,
                              void* d_out, int out_size, void* d_ws, size_t ws_size,
                              hipStream_t stream) {
  (void)in_sizes; (void)n_in; (void)out_size; (void)ws_size;
  const float* acts = (const float*)d_in[0];
  const float* u1   = (const float*)d_in[1];
  const float* u2   = (const float*)d_in[2];

  char* ws = (char*)d_ws;
  float*        elogit = (float*)(ws + OFF_ELOGIT);
  float*        prefW  = (float*)(ws + OFF_PREFW);
  unsigned int* fd     = (unsigned int*)(ws + OFF_FD);

  xrisk_precompute<<<1, BLOCKA, 0, stream>>>(acts, elogit, prefW);

  dim3 gridB(BATCH / 4 / 256, NCHUNK);   // (4, 250)
  xrisk_stream<<<gridB, 256, 0, stream>>>(u1, u2, elogit, fd);

  xrisk_reduce<<<1, 1024, 0, stream>>>((const unsigned char*)fd, prefW, (float*)d_out);
}